// GCN_48129403519136
// MI455X (gfx1250) — compile-verified
//
#include <hip/hip_runtime.h>

#define IN_F  128
#define HID_F 64
#define NCLS  16

typedef float v2f __attribute__((ext_vector_type(2)));
typedef float v8f __attribute__((ext_vector_type(8)));

// D = A(16x4 f32) * B(4x16 f32) + C(16x16 f32), wave32, fp32 precision kept.
__device__ __forceinline__ v8f wmma_f32x4(v2f a, v2f b, v8f c) {
  return __builtin_amdgcn_wmma_f32_16x16x4_f32(
      /*neg_a=*/false, a, /*neg_b=*/false, b,
      /*c_mod=*/(short)0, c, /*reuse_a=*/false, /*reuse_b=*/false);
}

// ---------------- elementwise / utility kernels ----------------

__global__ void k_fill(float* __restrict__ p, int n, float v) {
  int i = blockIdx.x * blockDim.x + threadIdx.x;
  if (i < n) p[i] = v;
}

__global__ void k_deg_atomic(const int* __restrict__ dst, float* __restrict__ deg, int e) {
  int i = blockIdx.x * blockDim.x + threadIdx.x;
  if (i < e) atomicAdd(&deg[dst[i]], 1.0f);
}

__global__ void k_rsqrt(float* __restrict__ p, int n) {
  int i = blockIdx.x * blockDim.x + threadIdx.x;
  if (i < n) p[i] = rsqrtf(p[i]);
}

// agg[i] = h[i] * dinv[node]^2   (self-loop contribution seeds the accumulator)
__global__ void k_selfloop(const float* __restrict__ h, const float* __restrict__ dinv,
                           float* __restrict__ agg, int total) {
  int i = blockIdx.x * blockDim.x + threadIdx.x;
  if (i >= total) return;
  float di = dinv[i >> 6];           // 64 feats per node
  agg[i] = h[i] * di * di;
}

// Edge-parallel scatter: 16 threads per edge, 4 floats each (float4 gather, 4 atomics).
__global__ void k_edge_scatter(const int* __restrict__ src, const int* __restrict__ dst,
                               const float* __restrict__ dinv, const float* __restrict__ h,
                               float* __restrict__ agg, int e) {
  int gid = blockIdx.x * blockDim.x + threadIdx.x;
  int edge = gid >> 4;
  if (edge >= e) return;
  int f = (gid & 15) << 2;
  int s = src[edge], d = dst[edge];
  float nrm = dinv[s] * dinv[d];
  const float4 hv = *(const float4*)(h + (size_t)s * HID_F + f);
  float* ap = agg + (size_t)d * HID_F + f;
  atomicAdd(ap + 0, hv.x * nrm);
  atomicAdd(ap + 1, hv.y * nrm);
  atomicAdd(ap + 2, hv.z * nrm);
  atomicAdd(ap + 3, hv.w * nrm);
}

__global__ void k_bias_relu(float* __restrict__ agg, const float* __restrict__ b, int total) {
  int i = blockIdx.x * blockDim.x + threadIdx.x;
  if (i >= total) return;
  agg[i] = fmaxf(agg[i] + b[i & (HID_F - 1)], 0.0f);
}

__global__ void k_pool_count(const int* __restrict__ dict, float* __restrict__ cnt, int n) {
  int i = blockIdx.x * blockDim.x + threadIdx.x;
  if (i < n) atomicAdd(&cnt[dict[i]], 1.0f);
}

__global__ void k_pool_sum(const int* __restrict__ dict, const float* __restrict__ h,
                           float* __restrict__ pool, int total) {
  int i = blockIdx.x * blockDim.x + threadIdx.x;
  if (i >= total) return;
  int node = i >> 6, f = i & (HID_F - 1);
  atomicAdd(&pool[(size_t)dict[node] * HID_F + f], h[i]);
}

// ---------------- fp32 WMMA GEMM:  O[n,64] = A[n,KD] @ W[KD,64] ----------------
// One wave computes a 16x64 output stripe: 4 accumulator tiles reuse the A fragment.
template <int KD>
__global__ void k_gemm_wmma(const float* __restrict__ A, const float* __restrict__ W,
                            float* __restrict__ O, int n) {
  const int wave = blockIdx.x * (blockDim.x >> 5) + (threadIdx.x >> 5);
  const int lane = threadIdx.x & 31;
  const int half = lane >> 4;          // K-pair / M-row-half selector
  const int lid  = lane & 15;
  const int row0 = wave << 4;
  if (row0 >= n) return;

  int m = row0 + lid;
  if (m >= n) m = n - 1;               // safe clamp; those output rows are unstored

  v8f acc0 = {}, acc1 = {}, acc2 = {}, acc3 = {};
  const float* ap = A + (size_t)m * KD + (half << 1);
  const float* wp = W + (size_t)(half << 1) * HID_F + lid;

  #pragma unroll 2
  for (int k = 0; k < KD; k += 4) {
    v2f a;                                   // A: M=lid(+half rows), K = k+2*half+{0,1}
    a.x = ap[0]; a.y = ap[1]; ap += 4;
    v2f b0, b1, b2, b3;                      // B: N = lid + 16*t, K rows split by half
    b0.x = wp[0];  b0.y = wp[HID_F + 0];
    b1.x = wp[16]; b1.y = wp[HID_F + 16];
    b2.x = wp[32]; b2.y = wp[HID_F + 32];
    b3.x = wp[48]; b3.y = wp[HID_F + 48];
    wp += 4 * HID_F;
    acc0 = wmma_f32x4(a, b0, acc0);
    acc1 = wmma_f32x4(a, b1, acc1);
    acc2 = wmma_f32x4(a, b2, acc2);
    acc3 = wmma_f32x4(a, b3, acc3);
  }

  const int rbase = row0 + (half << 3);      // C/D: lanes 0-15 -> M 0..7, 16-31 -> M 8..15
  #pragma unroll
  for (int r = 0; r < 8; ++r) {
    int row = rbase + r;
    if (row >= n) break;
    float* op = O + (size_t)row * HID_F + lid;
    op[0]  = acc0[r];
    op[16] = acc1[r];
    op[32] = acc2[r];
    op[48] = acc3[r];
  }
}

// ---------------- pooled mean -> FC(64x16) -> log_softmax ----------------
__global__ void k_fc_lsm(const float* __restrict__ pool, const float* __restrict__ cnt,
                         const float* __restrict__ Wfc, const float* __restrict__ bfc,
                         float* __restrict__ out, int n) {
  int i = blockIdx.x * blockDim.x + threadIdx.x;
  if (i >= n) return;
  float inv = 1.0f / fmaxf(cnt[i], 1.0f);
  float p[HID_F];
  #pragma unroll
  for (int k = 0; k < HID_F; ++k) p[k] = pool[(size_t)i * HID_F + k] * inv;
  float lg[NCLS];
  #pragma unroll
  for (int c = 0; c < NCLS; ++c) lg[c] = bfc[c];
  for (int k = 0; k < HID_F; ++k) {
    float pk = p[k];
    #pragma unroll
    for (int c = 0; c < NCLS; ++c) lg[c] += pk * Wfc[k * NCLS + c];
  }
  float mx = lg[0];
  #pragma unroll
  for (int c = 1; c < NCLS; ++c) mx = fmaxf(mx, lg[c]);
  float s = 0.0f;
  #pragma unroll
  for (int c = 0; c < NCLS; ++c) s += expf(lg[c] - mx);
  float lse = mx + logf(s);
  #pragma unroll
  for (int c = 0; c < NCLS; ++c) out[(size_t)i * NCLS + c] = lg[c] - lse;
}

// ---------------- launcher ----------------
static inline int cdiv(long a, int b) { return (int)((a + b - 1) / b); }

extern "C" void kernel_launch(void* const* d_in, const int* in_sizes, int n_in,
                              void* d_out, int out_size, void* d_ws, size_t ws_size,
                              hipStream_t stream) {
  const float* x    = (const float*)d_in[0];
  const int*   ei   = (const int*)  d_in[1];
  const int*   dict = (const int*)  d_in[2];
  const float* W1   = (const float*)d_in[3];
  const float* b1   = (const float*)d_in[4];
  const float* W2   = (const float*)d_in[5];
  const float* b2   = (const float*)d_in[6];
  const float* Wfc  = (const float*)d_in[7];
  const float* bfc  = (const float*)d_in[8];

  const int N = in_sizes[0] / IN_F;
  const int E = in_sizes[1] / 2;
  const int* srcv = ei;         // edge_index[0]
  const int* dstv = ei + E;     // edge_index[1]

  float* bufX = (float*)d_ws;                   // [N,64]
  float* bufY = bufX + (size_t)N * HID_F;       // [N,64]
  float* dinv = bufY + (size_t)N * HID_F;       // [N]
  float* cnt  = dinv + N;                       // [N]

  const int TPB = 256;
  const int NE  = N * HID_F;
  const int gemm_blocks = cdiv(cdiv(N, 16), 4); // 4 waves (128 thr) per block, 1 tile/wave

  // degree (self-loop seeds 1.0) -> dinv = rsqrt(deg)
  k_fill      <<<cdiv(N, TPB), TPB, 0, stream>>>(dinv, N, 1.0f);
  k_deg_atomic<<<cdiv(E, TPB), TPB, 0, stream>>>(dstv, dinv, E);
  k_rsqrt     <<<cdiv(N, TPB), TPB, 0, stream>>>(dinv, N);

  // layer 1: h = relu(norm_agg(x @ W1) + b1)
  k_gemm_wmma<IN_F><<<gemm_blocks, 128, 0, stream>>>(x, W1, bufX, N);
  k_selfloop   <<<cdiv(NE, TPB), TPB, 0, stream>>>(bufX, dinv, bufY, NE);
  k_edge_scatter<<<cdiv((long)E * 16, TPB), TPB, 0, stream>>>(srcv, dstv, dinv, bufX, bufY, E);
  k_bias_relu  <<<cdiv(NE, TPB), TPB, 0, stream>>>(bufY, b1, NE);

  // layer 2: h = relu(norm_agg(h @ W2) + b2)
  k_gemm_wmma<HID_F><<<gemm_blocks, 128, 0, stream>>>(bufY, W2, bufX, N);
  k_selfloop   <<<cdiv(NE, TPB), TPB, 0, stream>>>(bufX, dinv, bufY, NE);
  k_edge_scatter<<<cdiv((long)E * 16, TPB), TPB, 0, stream>>>(srcv, dstv, dinv, bufX, bufY, E);
  k_bias_relu  <<<cdiv(NE, TPB), TPB, 0, stream>>>(bufY, b2, NE);

  // global mean pool over dict_node, then FC + log_softmax
  k_fill      <<<cdiv(N, TPB), TPB, 0, stream>>>(cnt, N, 0.0f);
  k_fill      <<<cdiv(NE, TPB), TPB, 0, stream>>>(bufX, NE, 0.0f);
  k_pool_count<<<cdiv(N, TPB), TPB, 0, stream>>>(dict, cnt, N);
  k_pool_sum  <<<cdiv(NE, TPB), TPB, 0, stream>>>(dict, bufY, bufX, NE);
  k_fc_lsm    <<<cdiv(N, TPB), TPB, 0, stream>>>(bufX, cnt, Wfc, bfc, (float*)d_out, N);
}